// Decodering_model_33406255629189
// MI455X (gfx1250) — compile-verified
//
#include <hip/hip_runtime.h>
#include <cstdint>

// Problem constants (fixed by the reference).
#define M_ROWS   1024
#define N_COLS   2048
#define MAXDEG   16      // base row degree 8 + slack for empty-column fixups
#define NUM_ITERS 5

// ---------------------------------------------------------------------------
// Kernel 1: compact dense H [M,N] (int32 0/1) into per-row column lists.
// One wave32 per row; lanes scan 32 contiguous columns (128B coalesced reads),
// ballot + prefix popcount gives an order-preserving compaction, no atomics.
// H is the only real HBM traffic (8 MB) -> prefetch the stream.
// ---------------------------------------------------------------------------
__global__ void build_rows_kernel(const int* __restrict__ H,
                                  int* __restrict__ row_cols,
                                  int* __restrict__ row_deg) {
  const int row  = (int)((blockIdx.x * blockDim.x + threadIdx.x) >> 5);
  const int lane = (int)(threadIdx.x & 31u);
  if (row >= M_ROWS) return;

  const int* hrow = H + (size_t)row * N_COLS;
  int count = 0;
  for (int base = 0; base < N_COLS; base += 32) {
    if (base + 256 < N_COLS)
      __builtin_prefetch(hrow + base + 256 + lane, 0, 0);  // global_prefetch_b8
    const int h = hrow[base + lane];
    const unsigned mask = (unsigned)__ballot(h != 0);      // wave32: low 32 bits
    if (h != 0) {
      const int pos = count + __popc(mask & ((1u << lane) - 1u));
      if (pos < MAXDEG) row_cols[row * MAXDEG + pos] = base + lane;
    }
    count += __popc(mask);
  }
  if (lane == 0) row_deg[row] = (count < MAXDEG) ? count : MAXDEG;
}

// ---------------------------------------------------------------------------
// Kernel 2: LDS-resident normalized min-sum decode. One block per batch,
// one thread per check row.  soft_input is DMA'd into LDS and the posterior
// LLRs are DMA'd back out with the gfx1250 ASYNCcnt path
// (global_load_async_to_lds_b64 / global_store_async_from_lds_b64).
// ---------------------------------------------------------------------------
__launch_bounds__(M_ROWS)
__global__ void nms_decode_kernel(const float* __restrict__ soft_input,
                                  const float* __restrict__ check_weight,
                                  const int*   __restrict__ row_cols,
                                  const int*   __restrict__ row_deg,
                                  float*       __restrict__ soft_output) {
  extern __shared__ float smem[];
  float* s_in   = smem;            // [N_COLS] per-batch LLRs
  float* colsum = smem + N_COLS;   // [N_COLS] running column sums

  const int b   = (int)blockIdx.x;
  const int row = (int)threadIdx.x;          // blockDim.x == M_ROWS
  const int n2  = 2 * row;                   // this thread owns cols n2, n2+1

  // ---- stage soft_input[b, :] into LDS: one async B64 DMA per thread ----
  {
    const uint64_t g0 =
        (uint64_t)(uintptr_t)(soft_input + (size_t)b * N_COLS + n2);
    const uint32_t l0 = (uint32_t)(uintptr_t)(s_in + n2);
    asm volatile("global_load_async_to_lds_b64 %0, %1, off"
                 :: "v"(l0), "v"(g0) : "memory");
  }

  // Independent work overlapped with the async DMA:
  const float alpha = log1pf(expf(check_weight[0]));   // softplus (NMS factor)

  const int deg = row_deg[row];
  int   c [MAXDEG];
  float cv[MAXDEG];
#pragma unroll
  for (int k = 0; k < MAXDEG; ++k) {
    c[k]  = (k < deg) ? row_cols[row * MAXDEG + k] : 0;
    cv[k] = 0.0f;
  }

  asm volatile("s_wait_asynccnt 0" ::: "memory");  // my DMA landed in LDS
  __syncthreads();                                 // everyone's DMA visible

  for (int it = 0; it < NUM_ITERS; ++it) {
    // colsum = soft_input  (B64 LDS copy, 2 consecutive floats per thread)
    ((float2*)colsum)[row] = ((const float2*)s_in)[row];
    __syncthreads();

    // scatter-add cv into column sums (ds_add_f32 LDS float atomics)
#pragma unroll
    for (int k = 0; k < MAXDEG; ++k)
      if (k < deg) atomicAdd(&colsum[c[k]], cv[k]);
    __syncthreads();

    // variable-to-check messages on this row's edges
    float vc[MAXDEG];
#pragma unroll
    for (int k = 0; k < MAXDEG; ++k)
      if (k < deg) vc[k] = colsum[c[k]] - cv[k];

    // row min / second-min of clip(|vc|, 1e30) and sign product
    float m1 = 3.0e38f, m2 = 3.0e38f, sp = 1.0f;
#pragma unroll
    for (int k = 0; k < MAXDEG; ++k) {
      if (k < deg) {
        const float a = fminf(fabsf(vc[k]), 1.0e30f);
        if (a < m1)      { m2 = m1; m1 = a; }
        else if (a < m2) { m2 = a; }
        sp *= (vc[k] > 0.0f) ? 1.0f : ((vc[k] < 0.0f) ? -1.0f : 0.0f);
      }
    }

    // check-to-variable update: exclude-self min, product of other signs,
    // normalized by alpha.  (a > m1 ? m1 : m2) matches reference tie handling.
#pragma unroll
    for (int k = 0; k < MAXDEG; ++k) {
      if (k < deg) {
        const float a = fminf(fabsf(vc[k]), 1.0e30f);
        const float r = (a > m1) ? m1 : m2;
        const float s = (vc[k] > 0.0f) ? 1.0f : ((vc[k] < 0.0f) ? -1.0f : 0.0f);
        cv[k] = r * (sp * s) * alpha;
      }
    }
    __syncthreads();  // colsum reads done before next iteration overwrites it
  }

  // posterior LLRs: soft_input + sum_m cv  (accumulated in colsum)
  ((float2*)colsum)[row] = ((const float2*)s_in)[row];
  __syncthreads();
#pragma unroll
  for (int k = 0; k < MAXDEG; ++k)
    if (k < deg) atomicAdd(&colsum[c[k]], cv[k]);
  __syncthreads();

  // ---- write result straight LDS -> HBM: one async B64 DMA per thread ----
  {
    const uint64_t go =
        (uint64_t)(uintptr_t)(soft_output + (size_t)b * N_COLS + n2);
    const uint32_t lo = (uint32_t)(uintptr_t)(colsum + n2);
    asm volatile("global_store_async_from_lds_b64 %0, %1, off"
                 :: "v"(go), "v"(lo) : "memory");
    asm volatile("s_wait_asynccnt 0" ::: "memory");
  }
}

// ---------------------------------------------------------------------------
// Launch: inputs are {soft_input [B,N] f32, check_weight [1] f32, H [M,N] i32}
// ---------------------------------------------------------------------------
extern "C" void kernel_launch(void* const* d_in, const int* in_sizes, int n_in,
                              void* d_out, int out_size, void* d_ws, size_t ws_size,
                              hipStream_t stream) {
  const float* soft_input   = (const float*)d_in[0];
  const float* check_weight = (const float*)d_in[1];
  const int*   H            = (const int*)d_in[2];
  float*       soft_output  = (float*)d_out;

  const int B = in_sizes[0] / N_COLS;   // 8

  int* row_cols = (int*)d_ws;                      // M_ROWS * MAXDEG ints
  int* row_deg  = row_cols + M_ROWS * MAXDEG;      // M_ROWS ints  (~68 KB total)

  // 256 threads = 8 wave32 = 8 rows per block
  build_rows_kernel<<<dim3(M_ROWS / 8), dim3(256), 0, stream>>>(H, row_cols, row_deg);

  const size_t lds_bytes = (size_t)2 * N_COLS * sizeof(float);  // 16 KB
  nms_decode_kernel<<<dim3(B), dim3(M_ROWS), lds_bytes, stream>>>(
      soft_input, check_weight, row_cols, row_deg, soft_output);
}